// SparseMoE_46471546143557
// MI455X (gfx1250) — compile-verified
//
#include <hip/hip_runtime.h>
#include <hip/hip_bf16.h>
#include <math.h>

// Problem constants (from reference): B=4,S=2048,D=1024,E=8,TOPK=2,HID=4096
#define N_TOK 8192
#define DIM   1024
#define NEXP  8
#define NHID  4096
#define CAP   1280   // ceil(1.25 * 8192 / 8)
#define TOPKK 2

typedef __attribute__((ext_vector_type(2))) float v2f;
typedef __attribute__((ext_vector_type(8))) float v8f;
typedef __attribute__((ext_vector_type(4))) int   v4i;

// ---------------------------------------------------------------------------
// CDNA5 async global->LDS staging (GLOBAL_LOAD_ASYNC_TO_LDS_B128, ASYNCcnt).
// Builtin signature: (v4i addrspace(1)*, v4i addrspace(3)*, imm offset, imm cpol).
// Guarded so the kernel still compiles on toolchains without the builtin.
// ---------------------------------------------------------------------------
#if defined(__has_builtin)
#if __has_builtin(__builtin_amdgcn_global_load_async_to_lds_b128) && \
    __has_builtin(__builtin_amdgcn_s_wait_asynccnt)
#define MOE_ASYNC_LDS 1
#endif
#endif
#ifndef MOE_ASYNC_LDS
#define MOE_ASYNC_LDS 0
#endif

typedef __attribute__((address_space(1))) v4i global_v4i;
typedef __attribute__((address_space(3))) v4i lds_v4i;

__device__ __forceinline__ void copy16_to_lds(float* lds_dst, const float* gsrc) {
#if MOE_ASYNC_LDS
    __builtin_amdgcn_global_load_async_to_lds_b128(
        (global_v4i*)(gsrc), (lds_v4i*)(lds_dst), 0, 0);
#else
    *(float4*)lds_dst = *(const float4*)gsrc;
#endif
}

__device__ __forceinline__ void wait_stage_and_barrier() {
#if MOE_ASYNC_LDS
    __builtin_amdgcn_s_wait_asynccnt(0);
#endif
    __syncthreads();
}

__device__ __forceinline__ float gelu_exact(float x) {
    return 0.5f * x * (1.0f + erff(x * 0.70710678118654752440f));
}

// ---------------------------------------------------------------------------
// Kernel 1: router. One wave32 per token: logits = x[t] @ gate_w (D x 8),
// softmax over 8 experts, top-2 (strict > => lowest-index tie-break, matching
// jax.lax.top_k ordering).
// ---------------------------------------------------------------------------
__global__ __launch_bounds__(256) void moe_router(
    const float* __restrict__ x, const float* __restrict__ gw,
    int* __restrict__ topi, float* __restrict__ topv)
{
    const int lane = threadIdx.x & 31;
    const int wv   = threadIdx.x >> 5;         // wave32
    const int t    = blockIdx.x * 8 + wv;

    float acc[NEXP];
#pragma unroll
    for (int e = 0; e < NEXP; ++e) acc[e] = 0.f;

    const float* xr = x + (size_t)t * DIM;
    for (int i = lane; i < DIM; i += 32) {
        float xv = xr[i];
        const float4* g4 = (const float4*)(gw + (size_t)i * NEXP);
        float4 ga = g4[0], gb = g4[1];
        acc[0] += xv * ga.x; acc[1] += xv * ga.y;
        acc[2] += xv * ga.z; acc[3] += xv * ga.w;
        acc[4] += xv * gb.x; acc[5] += xv * gb.y;
        acc[6] += xv * gb.z; acc[7] += xv * gb.w;
    }
#pragma unroll
    for (int e = 0; e < NEXP; ++e) {
#pragma unroll
        for (int off = 16; off > 0; off >>= 1)
            acc[e] += __shfl_xor(acc[e], off, 32);
    }

    if (lane == 0) {
        float mx = acc[0];
#pragma unroll
        for (int e = 1; e < NEXP; ++e) mx = fmaxf(mx, acc[e]);
        float g[NEXP]; float s = 0.f;
#pragma unroll
        for (int e = 0; e < NEXP; ++e) { g[e] = __expf(acc[e] - mx); s += g[e]; }
        float inv = 1.f / s;
#pragma unroll
        for (int e = 0; e < NEXP; ++e) g[e] *= inv;
        int i0 = 0;
#pragma unroll
        for (int e = 1; e < NEXP; ++e) if (g[e] > g[i0]) i0 = e;
        int i1 = (i0 == 0) ? 1 : 0;
#pragma unroll
        for (int e = 0; e < NEXP; ++e) if (e != i1 && e != i0 && g[e] > g[i1]) i1 = e;
        topi[t * TOPKK + 0] = i0;  topv[t * TOPKK + 0] = g[i0];
        topi[t * TOPKK + 1] = i1;  topv[t * TOPKK + 1] = g[i1];
    }
}

// ---------------------------------------------------------------------------
// Kernel 2: token-order capacity assignment. One thread per expert does the
// sequential cumsum (exactly the reference's cumsum semantics). Unfilled
// capacity slots get token 0 / gate 0 (matching the reference's zero-init
// dispatch buffers). slot[t*2+k] = position in expert buffer, or -1 dropped.
// ---------------------------------------------------------------------------
__global__ void moe_route_capacity(
    const int* __restrict__ topi, const float* __restrict__ topv,
    int* __restrict__ buf_tok, float* __restrict__ buf_gate,
    int* __restrict__ slot)
{
    int e = threadIdx.x;
    if (e >= NEXP) return;
    int cnt = 0;
    for (int t = 0; t < N_TOK; ++t) {
#pragma unroll
        for (int k = 0; k < TOPKK; ++k) {
            if (topi[t * TOPKK + k] == e) {
                if (cnt < CAP) {
                    buf_tok[e * CAP + cnt]  = t;
                    buf_gate[e * CAP + cnt] = topv[t * TOPKK + k];
                    slot[t * TOPKK + k]     = cnt;
                } else {
                    slot[t * TOPKK + k] = -1;
                }
                cnt++;
            }
        }
    }
    for (int c = cnt; c < CAP; ++c) {
        buf_tok[e * CAP + c]  = 0;
        buf_gate[e * CAP + c] = 0.f;
    }
}

// ---------------------------------------------------------------------------
// Kernels 3/4: grouped GEMM on the f32 WMMA pipe (V_WMMA_F32_16X16X4_F32).
// Workgroup = 256 threads = 8 wave32s. Block tile: 128(M) x 128(N); each wave
// owns a 16x128 strip (8 x v8f accumulators = 64 VGPRs/lane). K is staged
// through double-buffered LDS in 32-wide chunks via
// GLOBAL_LOAD_ASYNC_TO_LDS_B128: the next chunk streams into LDS while the
// current chunk's 64 WMMAs execute; s_wait_asynccnt 0 + barrier swap buffers.
// Per kk-step, all 8 B fragments are preloaded into a v2f array (each as two
// loads off one base with DWORD offsets j*16 / j*16+LDB -> single
// ds_load_2addr_b32 into the fragment's adjacent even VGPR pair), so the
// scheduler can batch ~16 LDS loads ahead of the 8-WMMA burst with partial
// s_wait_dscnt waits instead of full drains.
// A-fragment layout (16x4 f32): lane l -> row M=l%16, VGPRs = K {2*(l/16), +1}.
// B-fragment (4x16):           lane l -> col N=l%16, VGPRs = K {2*(l/16), +1}.
// C/D (16x16): lane l -> N=l%16, VGPR v -> M = v + 8*(l/16).
// GATHER: A rows are x[buf_tok[...]] (fused dispatch). Epilogue: +bias,
// optional exact GELU (GEMM1), optional *gate (GEMM2).
// ---------------------------------------------------------------------------
template<int K_DIM, int N_DIM, bool DO_GELU, bool DO_GATE, bool GATHER>
__global__ __launch_bounds__(256) void moe_gemm(
    const float* __restrict__ A, const float* __restrict__ W,
    const float* __restrict__ bias, const float* __restrict__ gate,
    const int* __restrict__ tokmap, float* __restrict__ Cout)
{
    constexpr int MT = 128, NT = 128, KT = 32;
    constexpr int NJ  = NT / 16;  // 8 accumulator sets per wave
    constexpr int LDA = KT + 4;   // 36 floats: row stride 144B, 16B-aligned
    constexpr int LDB = NT + 4;   // 132 floats: row stride 528B, 16B-aligned
    __shared__ __align__(16) float lA[2 * MT * LDA];   // double-buffered
    __shared__ __align__(16) float lB[2 * KT * LDB];
    __shared__ int rowTok[MT];

    const int tid = threadIdx.x;
    const int e   = blockIdx.z;
    const int m0  = blockIdx.y * MT;
    const int n0  = blockIdx.x * NT;

    const float* We = W + (size_t)e * K_DIM * N_DIM;
    const float* Ae = GATHER ? A : (A + (size_t)e * CAP * K_DIM);

    if (GATHER) {
        if (tid < MT) rowTok[tid] = tokmap[e * CAP + m0 + tid];
        __syncthreads();
    }

    // Stage one K-chunk (A: 128x32, B: 32x128) into LDS buffer `buf`.
    auto stage = [&](int buf, int k0) {
        float* dA = &lA[buf * MT * LDA];
        float* dB = &lB[buf * KT * LDB];
#pragma unroll
        for (int i = 0; i < 4; ++i) {               // A: 1024 b128, 4/thread
            int idx = tid + i * 256;
            int r = idx >> 3, c4 = idx & 7;
            size_t arow = GATHER ? (size_t)rowTok[r] : (size_t)(m0 + r);
            copy16_to_lds(dA + r * LDA + c4 * 4,
                          Ae + arow * K_DIM + k0 + c4 * 4);
        }
#pragma unroll
        for (int i = 0; i < 4; ++i) {               // B: 1024 b128, 4/thread
            int idx = tid + i * 256;
            int r = idx >> 5, c4 = idx & 31;
            copy16_to_lds(dB + r * LDB + c4 * 4,
                          We + (size_t)(k0 + r) * N_DIM + n0 + c4 * 4);
        }
    };

    v8f acc[NJ] = {};
    const int lane = tid & 31;
    const int half = lane >> 4;
    const int lm   = lane & 15;
    const int wv   = tid >> 5;

    stage(0, 0);
    wait_stage_and_barrier();

    for (int k0 = 0; k0 < K_DIM; k0 += KT) {
        const int buf = (k0 / KT) & 1;
        if (k0 + KT < K_DIM) stage(buf ^ 1, k0 + KT);   // overlap with WMMAs

        const float* sA = &lA[buf * MT * LDA];
        const float* sB = &lB[buf * KT * LDB];
        const float* ap = sA + (wv * 16 + lm) * LDA + 2 * half;
        const float* bp = sB + 2 * half * LDB + lm;
#pragma unroll
        for (int kk = 0; kk < KT; kk += 4) {
            v2f a;
            a.x = ap[kk];
            a.y = ap[kk + 1];
            v2f b[NJ];
#pragma unroll
            for (int j = 0; j < NJ; ++j) {
                b[j].x = bp[kk * LDB + j * 16];        // row kk+2*half
                b[j].y = bp[kk * LDB + j * 16 + LDB];  // row kk+2*half+1
            }
#pragma unroll
            for (int j = 0; j < NJ; ++j)
                acc[j] = __builtin_amdgcn_wmma_f32_16x16x4_f32(
                    false, a, false, b[j], (short)0, acc[j], false, false);
        }
        wait_stage_and_barrier();   // next buffer landed; all reads of buf done
    }

    // Epilogue
#pragma unroll
    for (int j = 0; j < NJ; ++j) {
        int n = n0 + j * 16 + lm;
        float bv = bias[e * N_DIM + n];
#pragma unroll
        for (int v = 0; v < 8; ++v) {
            int m = wv * 16 + v + 8 * half;
            float val = acc[j][v] + bv;
            if (DO_GELU) val = gelu_exact(val);
            if (DO_GATE) val *= gate[e * CAP + m0 + m];
            Cout[((size_t)e * CAP + m0 + m) * N_DIM + n] = val;
        }
    }
}

// ---------------------------------------------------------------------------
// Kernel 5: per-token combine (gather, no atomics => deterministic). yw is
// already gate-weighted; dropped slots (-1) contribute 0. Also writes the
// aux_loss scalar (0) at the tail of d_out.
// ---------------------------------------------------------------------------
__global__ __launch_bounds__(256) void moe_combine(
    const int* __restrict__ topi, const int* __restrict__ slot,
    const float* __restrict__ yw, float* __restrict__ out)
{
    int t = blockIdx.x, tid = threadIdx.x;
    float4 s = make_float4(0.f, 0.f, 0.f, 0.f);
#pragma unroll
    for (int k = 0; k < TOPKK; ++k) {
        int sl = slot[t * TOPKK + k];
        if (sl >= 0) {
            int e = topi[t * TOPKK + k];
            float4 v = *(const float4*)(yw + ((size_t)e * CAP + sl) * DIM + tid * 4);
            s.x += v.x; s.y += v.y; s.z += v.z; s.w += v.w;
        }
    }
    *(float4*)(out + (size_t)t * DIM + tid * 4) = s;
    if (t == 0 && tid == 0) out[(size_t)N_TOK * DIM] = 0.f;  // aux_loss
}

// ---------------------------------------------------------------------------
extern "C" void kernel_launch(void* const* d_in, const int* in_sizes, int n_in,
                              void* d_out, int out_size, void* d_ws, size_t ws_size,
                              hipStream_t stream) {
    const float* x      = (const float*)d_in[0];   // (B,S,D)
    const float* gate_w = (const float*)d_in[1];   // (D,E)
    const float* w1     = (const float*)d_in[2];   // (E,D,HID)
    const float* b1     = (const float*)d_in[3];   // (E,HID)
    const float* w2     = (const float*)d_in[4];   // (E,HID,D)
    const float* b2     = (const float*)d_in[5];   // (E,D)
    float* out = (float*)d_out;

    char*  ws  = (char*)d_ws;
    size_t off = 0;
    auto alloc = [&](size_t bytes) -> void* {
        off = (off + 255) & ~(size_t)255;
        void* p = ws + off;
        off += bytes;
        return p;
    };
    int*   topi     = (int*)  alloc((size_t)N_TOK * TOPKK * sizeof(int));
    float* topv     = (float*)alloc((size_t)N_TOK * TOPKK * sizeof(float));
    int*   slot     = (int*)  alloc((size_t)N_TOK * TOPKK * sizeof(int));
    int*   buf_tok  = (int*)  alloc((size_t)NEXP * CAP * sizeof(int));
    float* buf_gate = (float*)alloc((size_t)NEXP * CAP * sizeof(float));
    float* h        = (float*)alloc((size_t)NEXP * CAP * NHID * sizeof(float)); // 160 MB
    float* yw       = (float*)alloc((size_t)NEXP * CAP * DIM  * sizeof(float)); // 40 MB
    (void)ws_size; (void)in_sizes; (void)n_in; (void)out_size;

    // 1) router
    moe_router<<<N_TOK / 8, 256, 0, stream>>>(x, gate_w, topi, topv);
    // 2) capacity assignment (token-order cumsum per expert)
    moe_route_capacity<<<1, 32, 0, stream>>>(topi, topv, buf_tok, buf_gate, slot);
    // 3) GEMM1: h = gelu(x[buf_tok] @ w1 + b1)   M=1280,K=1024,N=4096 per expert
    moe_gemm<DIM, NHID, true, false, true>
        <<<dim3(NHID / 128, CAP / 128, NEXP), 256, 0, stream>>>(
            x, w1, b1, nullptr, buf_tok, h);
    // 4) GEMM2: yw = (h @ w2 + b2) * gate        M=1280,K=4096,N=1024 per expert
    moe_gemm<NHID, DIM, false, true, false>
        <<<dim3(DIM / 128, CAP / 128, NEXP), 256, 0, stream>>>(
            h, w2, b2, buf_gate, nullptr, yw);
    // 5) combine back to token order (+ aux_loss scalar)
    moe_combine<<<N_TOK, 256, 0, stream>>>(topi, slot, yw, out);
}